// Shift3DLayer_55353538511077
// MI455X (gfx1250) — compile-verified
//
#include <hip/hip_runtime.h>
#include <cstdint>

// x: [B=8, C=4, D=64, H=256, W=256] float32
// Pure bandwidth-bound permutation: 1 GiB total traffic, ~46 us at 23.3 TB/s.
// No FLOPs -> no WMMA. CDNA5 path used: async global->LDS->global DMA
// (ASYNCcnt) for the straight-copy case, 128-bit coalesced VMEM elsewhere.

constexpr int Bn = 8, Cn = 4, Dn = 64, Hn = 256, Wn = 256;
constexpr int W4 = Wn / 4; // 64 float4 per row

__global__ __launch_bounds__(256) void shift3d_kernel(
    const float* __restrict__ x,
    const int* __restrict__ dims,
    const int* __restrict__ shifts,
    const int* __restrict__ signs,
    const int* __restrict__ apply_mask,
    float* __restrict__ out,
    int nq)
{
    __shared__ float4 stage[256]; // per-lane async staging slot (4 KB)

    const unsigned uq = blockIdx.x * blockDim.x + threadIdx.x;
    if (uq >= (unsigned)nq) return;

    // q layout (float4 units): [b:3][c:2][d:6][h:8][w4:6]  (2^25 total)
    const int w4 = uq & 63;
    const int h  = (uq >> 6) & 255;
    const int d  = (uq >> 14) & 63;
    const int c  = (uq >> 20) & 3;
    const int b  = uq >> 22;
    const int bc = b * Cn + c;

    const float4* __restrict__ x4 = (const float4*)x;
    float4* __restrict__ o4 = (float4*)out;

    const int apply = apply_mask[b];

    if (apply == 0) {
        // Straight copy: CDNA5 async DMA, global -> LDS -> global.
        // Per-lane 16B transfer; data never touches VGPRs.
        const uint32_t lds = (uint32_t)(uintptr_t)(&stage[threadIdx.x]);
        const float4* src = &x4[uq];
        float4* dst = &o4[uq];
        asm volatile("global_load_async_to_lds_b128 %0, %1, off"
                     :: "v"(lds), "v"(src) : "memory");
        asm volatile("s_wait_asynccnt 0" ::: "memory");
        asm volatile("global_store_async_from_lds_b128 %0, %1, off"
                     :: "v"(dst), "v"(lds) : "memory");
        // outstanding async store drained by s_endpgm's implicit wait-idle
        return;
    }

    const int dim   = dims[bc];
    const int shift = shifts[bc];
    const int sign  = signs[bc];     // -1 or +1
    const int s     = sign * shift;  // signed roll: out[i] = in[(i - s) mod n]

    float4 r;
    if (dim == 0) {
        // roll along D: whole row copies, aligned b128 load
        const bool pad = (sign == 1) ? (d < shift) : (d >= Dn - shift);
        if (pad) {
            r = make_float4(0.f, 0.f, 0.f, 0.f);
        } else {
            const int sd = (d - s + Dn) & (Dn - 1);
            const unsigned sq = uq + ((unsigned)(sd - d) << 14);
            r = x4[sq];
        }
    } else if (dim == 1) {
        // roll along H: whole row copies, aligned b128 load
        const bool pad = (sign == 1) ? (h < shift) : (h >= Hn - shift);
        if (pad) {
            r = make_float4(0.f, 0.f, 0.f, 0.f);
        } else {
            const int sh = (h - s + Hn) & (Hn - 1);
            const unsigned sq = uq + ((unsigned)(sh - h) << 6);
            r = x4[sq];
        }
    } else {
        // roll along W: element-granular shift inside the row
        const unsigned rowf = (uq >> 6) * (unsigned)Wn; // float index of row start
        const int w0 = w4 * 4;
        float v[4];
#pragma unroll
        for (int j = 0; j < 4; ++j) {
            const int w = w0 + j;
            const bool pad = (sign == 1) ? (w < shift) : (w >= Wn - shift);
            const int sw = (w - s + Wn) & (Wn - 1);
            v[j] = pad ? 0.f : x[rowf + (unsigned)sw];
        }
        r = make_float4(v[0], v[1], v[2], v[3]);
    }
    o4[uq] = r;
}

extern "C" void kernel_launch(void* const* d_in, const int* in_sizes, int n_in,
                              void* d_out, int out_size, void* d_ws, size_t ws_size,
                              hipStream_t stream) {
    const float* x      = (const float*)d_in[0];
    const int* dims     = (const int*)d_in[1];
    const int* shifts   = (const int*)d_in[2];
    const int* signs    = (const int*)d_in[3];
    const int* apply    = (const int*)d_in[4];
    float* out          = (float*)d_out;

    const int nq = out_size / 4;              // number of float4 outputs (2^25)
    const int blocks = (nq + 255) / 256;
    shift3d_kernel<<<blocks, 256, 0, stream>>>(x, dims, shifts, signs, apply, out, nq);
}